// ContrastiveLoss_76476187673027
// MI455X (gfx1250) — compile-verified
//
#include <hip/hip_runtime.h>
#include <hip/hip_bf16.h>
#include <math.h>
#include <stdint.h>

typedef __attribute__((ext_vector_type(2))) float v2f;
typedef __attribute__((ext_vector_type(8))) float v8f;

#define TEMP_F 0.07f
#define EPS_F 1e-12f
#define DDIM 128
#define NCHUNK 8          // j-sweep split per row tile (grid.y)

// ---------------- Kernel 1: label histogram (C=100 < 128 bins) ----------------
__global__ void k_hist(const int* __restrict__ labels, int B, int* __restrict__ counts) {
    __shared__ int h[128];
    int t = threadIdx.x;
    if (t < 128) h[t] = 0;
    __syncthreads();
    for (int i = t; i < B; i += blockDim.x)
        atomicAdd(&h[labels[i] & 127], 1);
    __syncthreads();
    if (t < 128) counts[t] = h[t];
}

// ------- Kernel 2: W[k,:] = (A[k,:] + (A_k^T C_{l_k}) / (2T^2)) / T -----------
__global__ void k_buildW(const float* __restrict__ F, const int* __restrict__ labels,
                         const float* __restrict__ cov, float* __restrict__ W) {
    __shared__ float a[DDIM];
    const int k = blockIdx.x;
    const int e = threadIdx.x;                      // 0..127 (output column)
    a[e] = F[(size_t)k * DDIM + e];
    __syncthreads();
    const float* Cm = cov + (size_t)(labels[k] & 127) * DDIM * DDIM;
    float v = 0.f;
#pragma unroll 8
    for (int d = 0; d < DDIM; ++d)
        v = fmaf(a[d], Cm[(size_t)d * DDIM + e], v);   // coalesced across threads
    const float inv2T2 = 1.0f / (2.0f * TEMP_F * TEMP_F);
    const float invT   = 1.0f / TEMP_F;
    W[(size_t)k * DDIM + e] = (a[e] + v * inv2T2) * invT;
}

// ---- Kernel 3: logits = W @ A^T via V_WMMA_F32_16X16X4_F32, fused online -----
// grid = (B/16 row tiles) x (NCHUNK j-chunks); 128 threads = 4 waves.
// The 16x128 W row tile is staged into LDS with the gfx1250 async-DMA path
// (GLOBAL_LOAD_ASYNC_TO_LDS_B128, ASYNCcnt). Each wave sweeps 1/4 of its
// chunk's j-tiles, FOUR tiles at a time (4 independent WMMA accumulator
// chains -> 4-load clauses, 4 back-to-back WMMAs), maintaining per-row online
// (max, sumexp, sum). Wave partials merge via shfl_xor + LDS; the block
// writes per-(rowTile,chunk) partials to workspace.
__global__ void k_logits(const float* __restrict__ W, const float* __restrict__ F,
                         float* __restrict__ pmg, float* __restrict__ psg,
                         float* __restrict__ ptg, int B) {
    __shared__ float lw[16 * DDIM];                 // 8 KB W tile
    __shared__ float pm[4][16], ps[4][16], pt[4][16];

    const int lane = threadIdx.x & 31;
    const int wave = threadIdx.x >> 5;
    const int half = lane >> 4;       // which half of the wave
    const int lr   = lane & 15;       // row (A operand) / col (B operand) in tile
    const int rowBase = blockIdx.x * 16;
    const int chunk   = blockIdx.y;

    // ---- async stage of the 8KB W row tile into LDS (ASYNCcnt path) ----
    // per lane: 16B at lds[t*16 + off] <- W[rowBase*128 + t*4 + off/4]
    // 128 threads x 4 ops x 16B = 8KB; IOFFSET applies to both sides.
    {
        unsigned ldsAddr  = (unsigned)(size_t)(&lw[threadIdx.x * 4]);
        const float* gptr = W + (size_t)rowBase * DDIM + threadIdx.x * 4;
        asm volatile(
            "global_load_async_to_lds_b128 %0, %1, off\n\t"
            "global_load_async_to_lds_b128 %0, %1, off offset:2048\n\t"
            "global_load_async_to_lds_b128 %0, %1, off offset:4096\n\t"
            "global_load_async_to_lds_b128 %0, %1, off offset:6144"
            :: "v"(ldsAddr), "v"(gptr)
            : "memory");
        asm volatile("s_wait_asynccnt 0x0" ::: "memory");
    }
    __syncthreads();

    // A-operand per lane: lw[lr][k0 + 2*half + {0,1}]  (ISA 16x4 f32 layout)
    const float* wl = lw + lr * DDIM + 2 * half;

    const int numJT   = B / 16;            // 256
    const int jtChunk = numJT / NCHUNK;    // 32
    const int perWave = jtChunk / 4;       // 8
    const int jtBase  = chunk * jtChunk + wave * perWave;

    float m[8], s[8], t[8];
#pragma unroll
    for (int g = 0; g < 8; ++g) { m[g] = -INFINITY; s[g] = 0.f; t[g] = 0.f; }

    for (int jp = 0; jp < perWave; jp += 4) {
        const int jt0 = jtBase + jp;
        // B-operand: A[jt*16+lr][k0 + 2*half + {0,1}]  (4x16 layout, symmetric)
        const float* ap0 = F + (size_t)(jt0 * 16 + lr) * DDIM + 2 * half;
        const float* ap1 = ap0 + 16 * DDIM;
        const float* ap2 = ap0 + 32 * DDIM;
        const float* ap3 = ap0 + 48 * DDIM;
        v8f acc0 = {0.f,0.f,0.f,0.f,0.f,0.f,0.f,0.f};
        v8f acc1 = {0.f,0.f,0.f,0.f,0.f,0.f,0.f,0.f};
        v8f acc2 = {0.f,0.f,0.f,0.f,0.f,0.f,0.f,0.f};
        v8f acc3 = {0.f,0.f,0.f,0.f,0.f,0.f,0.f,0.f};
#pragma unroll
        for (int k0 = 0; k0 < DDIM; k0 += 4) {
            v2f av = *(const v2f*)(wl + k0);
            v2f b0 = *(const v2f*)(ap0 + k0);
            v2f b1 = *(const v2f*)(ap1 + k0);
            v2f b2 = *(const v2f*)(ap2 + k0);
            v2f b3 = *(const v2f*)(ap3 + k0);
            acc0 = __builtin_amdgcn_wmma_f32_16x16x4_f32(
                false, av, false, b0, (short)0, acc0, false, false);
            acc1 = __builtin_amdgcn_wmma_f32_16x16x4_f32(
                false, av, false, b1, (short)0, acc1, false, false);
            acc2 = __builtin_amdgcn_wmma_f32_16x16x4_f32(
                false, av, false, b2, (short)0, acc2, false, false);
            acc3 = __builtin_amdgcn_wmma_f32_16x16x4_f32(
                false, av, false, b3, (short)0, acc3, false, false);
        }
        // lane holds 4 columns for rows g + 8*half : online softmax update
#pragma unroll
        for (int g = 0; g < 8; ++g) {
            float x0 = acc0[g], x1 = acc1[g], x2 = acc2[g], x3 = acc3[g];
            t[g] += (x0 + x1) + (x2 + x3);
            float mn = fmaxf(fmaxf(m[g], fmaxf(x0, x1)), fmaxf(x2, x3));
            s[g] = s[g] * __expf(m[g] - mn)
                 + __expf(x0 - mn) + __expf(x1 - mn)
                 + __expf(x2 - mn) + __expf(x3 - mn);
            m[g] = mn;
        }
    }

    // merge the 16 column-lanes of each half (xor 1,2,4,8 stays within half)
#pragma unroll
    for (int off = 1; off < 16; off <<= 1) {
#pragma unroll
        for (int g = 0; g < 8; ++g) {
            float mo = __shfl_xor(m[g], off, 32);
            float so = __shfl_xor(s[g], off, 32);
            float to = __shfl_xor(t[g], off, 32);
            float mn = fmaxf(m[g], mo);
            s[g] = s[g] * __expf(m[g] - mn) + so * __expf(mo - mn);
            m[g] = mn;
            t[g] += to;
        }
    }

    if (lr == 0) {                 // lanes 0 and 16 carry rows 0-7 / 8-15
#pragma unroll
        for (int g = 0; g < 8; ++g) {
            int r = g + 8 * half;
            pm[wave][r] = m[g]; ps[wave][r] = s[g]; pt[wave][r] = t[g];
        }
    }
    __syncthreads();

    if (threadIdx.x < 16) {        // merge the 4 wave partials; write chunk partial
        int r = threadIdx.x;
        float M = pm[0][r], S = ps[0][r], T = pt[0][r];
        for (int wv = 1; wv < 4; ++wv) {
            float mo = pm[wv][r], so = ps[wv][r], to = pt[wv][r];
            float mn = fmaxf(M, mo);
            S = S * __expf(M - mn) + so * __expf(mo - mn);
            M = mn;
            T += to;
        }
        size_t idx = (size_t)chunk * B + rowBase + r;
        pmg[idx] = M; psg[idx] = S; ptg[idx] = T;
    }
}

// ---- Kernel 4: merge chunk partials per row, then
//      loss = -(1/B^2) Σ_k (ms_k/(ms_k+eps)) (t_k - B m_k - B log(s_k+eps))
__global__ void k_final(const float* __restrict__ pmg, const float* __restrict__ psg,
                        const float* __restrict__ ptg, const int* __restrict__ labels,
                        const int* __restrict__ counts, int B, float* __restrict__ out) {
    __shared__ float red[256];
    float local = 0.f;
    for (int k = threadIdx.x; k < B; k += 256) {
        float M = -INFINITY, S = 0.f, T = 0.f;
        for (int c = 0; c < NCHUNK; ++c) {
            size_t idx = (size_t)c * B + k;
            float mo = pmg[idx], so = psg[idx], to = ptg[idx];
            float mn = fmaxf(M, mo);
            S = S * __expf(M - mn) + so * __expf(mo - mn);
            M = mn;
            T += to;
        }
        float ms = (float)counts[labels[k] & 127];
        float wk = ms / (ms + EPS_F);
        float r  = T - (float)B * M - (float)B * logf(S + EPS_F);
        local -= wk * r;
    }
    red[threadIdx.x] = local;
    __syncthreads();
    for (int off = 128; off > 0; off >>= 1) {
        if ((int)threadIdx.x < off) red[threadIdx.x] += red[threadIdx.x + off];
        __syncthreads();
    }
    if (threadIdx.x == 0) out[0] = red[0] / ((float)B * (float)B);
}

extern "C" void kernel_launch(void* const* d_in, const int* in_sizes, int n_in,
                              void* d_out, int out_size, void* d_ws, size_t ws_size,
                              hipStream_t stream) {
    (void)n_in; (void)out_size; (void)ws_size;
    const float* F      = (const float*)d_in[0];   // [B,1,128] f32
    const int*   labels = (const int*)d_in[1];     // [B] int
    const float* cov    = (const float*)d_in[2];   // [C,128,128] f32
    const int B = in_sizes[1];                     // 4096

    // workspace layout
    char* ws = (char*)d_ws;
    float* W      = (float*)ws;                              // B*128 f32
    size_t off    = (size_t)B * DDIM * sizeof(float);
    int*   counts = (int*)(ws + off);  off += 128 * sizeof(int);
    float* pmg    = (float*)(ws + off); off += (size_t)NCHUNK * B * sizeof(float);
    float* psg    = (float*)(ws + off); off += (size_t)NCHUNK * B * sizeof(float);
    float* ptg    = (float*)(ws + off);

    k_hist  <<<1, 256, 0, stream>>>(labels, B, counts);
    k_buildW<<<B, 128, 0, stream>>>(F, labels, cov, W);
    dim3 g3(B / 16, NCHUNK);
    k_logits<<<g3, 128, 0, stream>>>(W, F, pmg, psg, ptg, B);
    k_final <<<1, 256, 0, stream>>>(pmg, psg, ptg, labels, counts, B, (float*)d_out);
}